// MessageBlock_13005160972655
// MI455X (gfx1250) — compile-verified
//
#include <hip/hip_runtime.h>
#include <hip/hip_bf16.h>
#include <math.h>

typedef __attribute__((ext_vector_type(2))) float v2f;
typedef __attribute__((ext_vector_type(8))) float v8f;
typedef __attribute__((ext_vector_type(4))) unsigned int u32x4;
typedef __attribute__((ext_vector_type(8))) int i32x8;
typedef __attribute__((ext_vector_type(4))) int i32x4;

#define FEAT      128
#define N_RBF     20
#define N_GRAPHS  128
#define ATOMS     64
#define N_ATOMS   (N_GRAPHS * ATOMS)      // 8192
#define N_EDGES   (N_ATOMS * 32)          // 262144
#define OUT_S_LEN (N_ATOMS * FEAT)        // 1048576

__device__ __forceinline__ void atomicAddF(float* p, float v) {
    __hip_atomic_fetch_add(p, v, __ATOMIC_RELAXED, __HIP_MEMORY_SCOPE_AGENT);
}

__device__ __forceinline__ float silu_f(float x) {
    return x / (1.0f + expf(-x));
}

// ---------------------------------------------------------------------------
// TDM: 1D tensor_load_to_lds. Builds a D# per CDNA5 ISA §8:
//   group0: [1:0]=count=1, [63:32]=lds_addr, [120:64]=global_addr, [127:126]=type=2
//   group1: [17:16]=data_size(2 -> 4B), [79:48]=tensor_dim0, [111:80]=tensor_dim1=1,
//           [127:112]=tile_dim0, [143:128]=tile_dim1=1, [207:160]=tensor_dim0_stride
//   groups 2/3 (+trailing group on 6-arg toolchain): zero (<=2D tensor)
// Issued by one wave; completion via TENSORcnt.
// ---------------------------------------------------------------------------
__device__ __forceinline__ void tdm_load_1d(const void* gptr,
                                            unsigned int lds_byte_addr,
                                            unsigned int nelem) // f32 elements, < 65536
{
    const unsigned long long ga = (unsigned long long)(uintptr_t)gptr;
    u32x4 g0;
    g0[0] = 1u;                                          // count=1, user descriptor
    g0[1] = lds_byte_addr;                               // LDS byte address
    g0[2] = (unsigned int)(ga & 0xFFFFFFFFu);            // global_addr[31:0]
    g0[3] = (unsigned int)((ga >> 32) & 0x01FFFFFFu)     // global_addr[56:32]
          | (2u << 30);                                  // type = 2 ("image")
    i32x8 g1;
    g1[0] = (int)(2u << 16);                             // data_size = 4 bytes
    g1[1] = (int)((nelem & 0xFFFFu) << 16);              // tensor_dim0[15:0]
    g1[2] = (int)((nelem >> 16) | (1u << 16));           // tensor_dim0[31:16], tensor_dim1=1
    g1[3] = (int)((nelem & 0xFFFFu) << 16);              // tile_dim0 = nelem
    g1[4] = 1;                                           // tile_dim1 = 1, tile_dim2 = 0
    g1[5] = (int)nelem;                                  // tensor_dim0_stride[31:0]
    g1[6] = 0;                                           // stride0 hi / stride1 lo
    g1[7] = 0;
    const i32x4 z4 = {0, 0, 0, 0};
    const i32x8 z8 = {0, 0, 0, 0, 0, 0, 0, 0};
    __builtin_amdgcn_tensor_load_to_lds(g0, g1, z4, z4, z8, 0);
}

// ---------------------------------------------------------------------------
// Generic fp32 WMMA GEMM: C[M,N] = act(A[M,K] @ B[K,N] + bias[N])
// One wave computes one 16x16 output tile via V_WMMA_F32_16X16X4_F32.
// ---------------------------------------------------------------------------
__global__ void wmma_gemm_kernel(const float* __restrict__ A,
                                 const float* __restrict__ B,
                                 const float* __restrict__ bias,
                                 float* __restrict__ C,
                                 int M, int N, int K, int act)
{
    const int wave = threadIdx.x >> 5;
    const int lane = threadIdx.x & 31;
    const int half = lane >> 4;     // 0: K 0..1 | rows v ; 1: K 2..3 | rows v+8
    const int lm   = lane & 15;

    const int tilesN = N >> 4;
    const int totalTiles = (M >> 4) * tilesN;
    const int w = blockIdx.x * (blockDim.x >> 5) + wave;
    if (w >= totalTiles) return;                     // wave-uniform

    const int tm = w / tilesN;
    const int tn = w - tm * tilesN;

    const float* Ap = A + (size_t)(tm * 16 + lm) * K + 2 * half;
    const float* Bp = B + (size_t)(2 * half) * N + tn * 16 + lm;

    v8f acc = {};
    for (int k0 = 0; k0 < K; k0 += 4) {
        v2f a, b;
        a.x = Ap[k0];
        a.y = Ap[k0 + 1];
        b.x = Bp[(size_t)k0 * N];
        b.y = Bp[(size_t)(k0 + 1) * N];
        acc = __builtin_amdgcn_wmma_f32_16x16x4_f32(false, a, false, b,
                                                    (short)0, acc, false, false);
    }

    const int n = tn * 16 + lm;
    const float bn = bias[n];
#pragma unroll
    for (int v = 0; v < 8; ++v) {
        const int m = tm * 16 + v + 8 * half;
        float x = acc[v] + bn;
        if (act) x = silu_f(x);
        C[(size_t)m * N + n] = x;
    }
}

// ---------------------------------------------------------------------------
// Edge message kernel: one edge per wave, 4 features per lane.
// Wr/br staged into LDS by the Tensor Data Mover (one descriptor per array,
// issued by wave 0, TENSORcnt-tracked), amortized over 8 edges per wave.
// ---------------------------------------------------------------------------
__global__ void edge_kernel(const float* __restrict__ phi,
                            const float* __restrict__ v_j,
                            const float* __restrict__ r_ij,
                            const int*   __restrict__ nbrs,
                            const float* __restrict__ Wr,
                            const float* __restrict__ br,
                            float* __restrict__ out)
{
    __shared__ float sWr[N_RBF * 3 * FEAT];   // 20*384 = 30720 B
    __shared__ float sbr[3 * FEAT];           // 1536 B

    if (threadIdx.x < 32) {
        tdm_load_1d(Wr, (unsigned int)(uintptr_t)&sWr[0], N_RBF * 3 * FEAT);
        tdm_load_1d(br, (unsigned int)(uintptr_t)&sbr[0], 3 * FEAT);
        __builtin_amdgcn_s_wait_tensorcnt(0);
    }
    __syncthreads();

    const int wave = threadIdx.x >> 5;
    const int lane = threadIdx.x & 31;
    const int gw   = blockIdx.x * (blockDim.x >> 5) + wave;
    const int nw   = gridDim.x * (blockDim.x >> 5);

    float* out_s = out;
    float* out_v = out + OUT_S_LEN;
    const float PI_OVER_CUT = 3.14159265358979323846f / 5.0f;
    const int fb = lane * 4;

    for (int e = gw; e < N_EDGES; e += nw) {
        if (e + nw < N_EDGES) {                       // global_prefetch_b8
            __builtin_prefetch(&r_ij[3 * (e + nw)], 0, 1);
            __builtin_prefetch(&nbrs[2 * (e + nw)], 0, 1);
        }

        const int i = nbrs[2 * e + 0];
        const int j = nbrs[2 * e + 1];

        const float rx = r_ij[3 * e + 0];
        const float ry = r_ij[3 * e + 1];
        const float rz = r_ij[3 * e + 2];
        const float dist = sqrtf(rx * rx + ry * ry + rz * rz);
        const float invd = 1.0f / dist;
        const float ux = rx * invd, uy = ry * invd, uz = rz * invd;
        const float env = (dist < 5.0f)
                            ? 0.5f * (cosf(PI_OVER_CUT * dist) + 1.0f)
                            : 0.0f;

        float rbf[N_RBF];
#pragma unroll
        for (int r = 0; r < N_RBF; ++r)
            rbf[r] = sinf((float)(r + 1) * PI_OVER_CUT * dist) * invd;

        float invv[3][4];
#pragma unroll
        for (int c = 0; c < 3; ++c) {
            const int col = c * FEAT + fb;
            float a0 = sbr[col + 0];
            float a1 = sbr[col + 1];
            float a2 = sbr[col + 2];
            float a3 = sbr[col + 3];
#pragma unroll
            for (int r = 0; r < N_RBF; ++r) {
                const float* row = sWr + r * (3 * FEAT) + col;
                const float rv = rbf[r];
                a0 = fmaf(rv, row[0], a0);
                a1 = fmaf(rv, row[1], a1);
                a2 = fmaf(rv, row[2], a2);
                a3 = fmaf(rv, row[3], a3);
            }
            const float4 p =
                *reinterpret_cast<const float4*>(phi + (size_t)j * (3 * FEAT) + col);
            invv[c][0] = p.x * a0 * env;
            invv[c][1] = p.y * a1 * env;
            invv[c][2] = p.z * a2 * env;
            invv[c][3] = p.w * a3 * env;
        }

#pragma unroll
        for (int u = 0; u < 4; ++u) {
            const int f = fb + u;
            atomicAddF(out_s + (size_t)i * FEAT + f, invv[1][u]);

            const float* vp = v_j + ((size_t)j * FEAT + f) * 3;
            const float s0v = invv[0][u];
            const float s2v = invv[2][u];
            float* ov = out_v + ((size_t)i * FEAT + f) * 3;
            atomicAddF(ov + 0, fmaf(s2v, ux, s0v * vp[0]));
            atomicAddF(ov + 1, fmaf(s2v, uy, s0v * vp[1]));
            atomicAddF(ov + 2, fmaf(s2v, uz, s0v * vp[2]));
        }
    }
}

// ---------------------------------------------------------------------------
// Per-graph attention: scores = Q@K^T/sqrt(F); softmax; att = P@V.
// ---------------------------------------------------------------------------
__global__ void attention_kernel(const float* __restrict__ qkv,
                                 float* __restrict__ out_s)
{
    __shared__ float P[ATOMS * ATOMS];        // 64*64*4 = 16 KB
    const int g    = blockIdx.x;
    const int tid  = threadIdx.x;
    const int wave = tid >> 5;
    const int lane = tid & 31;
    const int half = lane >> 4;
    const int lm   = lane & 15;
    const float* base = qkv + (size_t)g * ATOMS * (3 * FEAT);
    const float scale = 0.08838834764831845f;     // 1/sqrt(128)

    // ---- scores = Q @ K^T : 4x4 tiles of 16x16, 2 tiles per wave ----
    for (int t = wave; t < 16; t += 8) {
        const int tm = t >> 2, tn = t & 3;
        const float* Ap = base + (size_t)(tm * 16 + lm) * 384 + 2 * half;        // Q
        const float* Bp = base + (size_t)(tn * 16 + lm) * 384 + FEAT + 2 * half; // K^T col = K row
        v8f acc = {};
        for (int k0 = 0; k0 < FEAT; k0 += 4) {
            v2f a, b;
            a.x = Ap[k0]; a.y = Ap[k0 + 1];
            b.x = Bp[k0]; b.y = Bp[k0 + 1];
            acc = __builtin_amdgcn_wmma_f32_16x16x4_f32(false, a, false, b,
                                                        (short)0, acc, false, false);
        }
#pragma unroll
        for (int v = 0; v < 8; ++v)
            P[(tm * 16 + v + 8 * half) * ATOMS + tn * 16 + lm] = acc[v] * scale;
    }
    __syncthreads();

    // ---- row softmax (64 rows, thread per row) ----
    if (tid < ATOMS) {
        float mx = -1e30f;
        for (int k = 0; k < ATOMS; ++k) mx = fmaxf(mx, P[tid * ATOMS + k]);
        float s = 0.0f;
        for (int k = 0; k < ATOMS; ++k) {
            const float ev = expf(P[tid * ATOMS + k] - mx);
            P[tid * ATOMS + k] = ev;
            s += ev;
        }
        const float inv = 1.0f / s;
        for (int k = 0; k < ATOMS; ++k) P[tid * ATOMS + k] *= inv;
    }
    __syncthreads();

    // ---- att = P @ V : 4x8 tiles of 16x16, 4 tiles per wave; += into out ----
    for (int t = wave; t < 32; t += 8) {
        const int tm = t >> 3, tn = t & 7;
        const float* Ap = P + (tm * 16 + lm) * ATOMS + 2 * half;
        const float* Bp = base + (size_t)(2 * half) * 384 + 2 * FEAT + tn * 16 + lm;
        v8f acc = {};
        for (int k0 = 0; k0 < ATOMS; k0 += 4) {
            v2f a, b;
            a.x = Ap[k0]; a.y = Ap[k0 + 1];
            b.x = Bp[(size_t)k0 * 384];
            b.y = Bp[(size_t)(k0 + 1) * 384];
            acc = __builtin_amdgcn_wmma_f32_16x16x4_f32(false, a, false, b,
                                                        (short)0, acc, false, false);
        }
#pragma unroll
        for (int v = 0; v < 8; ++v) {
            const int m = g * ATOMS + tm * 16 + v + 8 * half;
            const int n = tn * 16 + lm;
            out_s[(size_t)m * FEAT + n] += acc[v];
        }
    }
}

// ---------------------------------------------------------------------------
extern "C" void kernel_launch(void* const* d_in, const int* in_sizes, int n_in,
                              void* d_out, int out_size, void* d_ws, size_t ws_size,
                              hipStream_t stream)
{
    const float* s_j  = (const float*)d_in[0];
    const float* v_j  = (const float*)d_in[1];
    const float* r_ij = (const float*)d_in[2];
    const int*   nbrs = (const int*)d_in[3];
    /* d_in[4] = num_atoms (constant 64 per graph) unused */
    const float* W1 = (const float*)d_in[5];
    const float* b1 = (const float*)d_in[6];
    const float* W2 = (const float*)d_in[7];
    const float* b2 = (const float*)d_in[8];
    const float* Wr = (const float*)d_in[9];
    const float* br = (const float*)d_in[10];
    const float* Wd = (const float*)d_in[11];
    const float* bd = (const float*)d_in[12];

    float* out = (float*)d_out;
    float* phi = (float*)d_ws;                         // 8192*384 f32
    float* qkv = phi + (size_t)N_ATOMS * 3 * FEAT;     // 8192*384 f32
    float* H   = qkv + (size_t)N_ATOMS * 3 * FEAT;     // 8192*128 f32

    // zero outputs (atomics + accumulation target)
    (void)hipMemsetAsync(d_out, 0, (size_t)out_size * sizeof(float), stream);

    // H = silu(s_j @ W1 + b1)         tiles = 512*8  = 4096 -> 512 blocks
    wmma_gemm_kernel<<<512, 256, 0, stream>>>(s_j, W1, b1, H,
                                              N_ATOMS, FEAT, FEAT, 1);
    // phi = H @ W2 + b2               tiles = 512*24 = 12288 -> 1536 blocks
    wmma_gemm_kernel<<<1536, 256, 0, stream>>>(H, W2, b2, phi,
                                               N_ATOMS, 3 * FEAT, FEAT, 0);
    // qkv = s_j @ Wd + bd             tiles = 512*24 = 12288 -> 1536 blocks
    wmma_gemm_kernel<<<1536, 256, 0, stream>>>(s_j, Wd, bd, qkv,
                                               N_ATOMS, 3 * FEAT, FEAT, 0);
    // edge messages + segment-sum via L2 atomics
    edge_kernel<<<4096, 256, 0, stream>>>(phi, v_j, r_ij, nbrs, Wr, br, out);
    // per-graph attention, accumulate into delta_s region
    attention_kernel<<<N_GRAPHS, 256, 0, stream>>>(qkv, out);
}